// MultiChannelSpikingAttention_90314572300875
// MI455X (gfx1250) — compile-verified
//
#include <hip/hip_runtime.h>
#include <math.h>

#define NB   4096          // batch rows
#define NT   8192          // time steps
#define CHK  64            // columns per staged chunk
#define NCHK (NT / CHK)    // 128 chunks
#define NBUF 4             // LDS ring depth (3 chunks in flight, <=63 asynccnt)
#define IPC  16            // async b128 issues per chunk (2 rows per issue)
#define WPR  (NT / 32)     // 256 packed words per row

// XOR swizzle for kernel-2 sal[] LDS indexing: bijective on [0,NT), kills
// bank conflicts in the expand / top-k passes, conflict-free in output pass.
#define SWZ(t) ((t) ^ (((t) >> 5) & 31))

// ---------------------------------------------------------------------------
// Kernel 1: LIF scan. 1 wave per workgroup; lane = one row of one channel.
// Tiles of 32 rows x 64 cols are staged into a 4-deep LDS ring with CDNA5
// async tensor-path loads (global_load_async_to_lds_b128, ASYNCcnt).
// Each issue: lanes 0-15 cover row 2j, lanes 16-31 row 2j+1 (2 x 256B
// coalesced). 16B-block rotation swizzle -> DMA stores and per-step
// lane-per-row ds_load_b128 reads are both conflict-free (2-way = b128 min).
// Spikes are bit-packed 32-per-u32 into the workspace.
// ---------------------------------------------------------------------------

__device__ __forceinline__ void issue_chunk_async(const float* src, int row0,
                                                  int lane, unsigned shbase,
                                                  int m) {
  const unsigned bufoff = (unsigned)(m & (NBUF - 1)) * (32u * CHK * 4u);
  const size_t t0 = (size_t)m * CHK;
  const int half = lane >> 4;        // 0 or 1: which row within the issue
  const int seg  = lane & 15;        // 16B segment within the row
#pragma unroll 4
  for (int j = 0; j < IPC; ++j) {
    const int r = 2 * j + half;      // local row 0..31
    const float* gp = src + (size_t)(row0 + r) * NT + t0 + 4 * seg;
    unsigned long long gaddr = (unsigned long long)(size_t)gp;
    const unsigned pos = (unsigned)((seg + r) & 15); // rotation swizzle
    unsigned ldsoff = shbase + bufoff + (unsigned)r * (CHK * 4u) + pos * 16u;
    asm volatile("global_load_async_to_lds_b128 %0, %1, off"
                 :: "v"(ldsoff), "v"(gaddr) : "memory");
  }
}

__global__ void __launch_bounds__(32)
lif_scan_kernel(const float* __restrict__ amp,
                const float* __restrict__ pitch,
                const float* __restrict__ boundary,
                const float* __restrict__ decay,
                unsigned* __restrict__ spk) {
  __shared__ __align__(16) float sh[NBUF * 32 * CHK]; // 32 KB ring

  const int lane = (int)threadIdx.x;      // 0..31, lane == local row for scan
  const int bid  = (int)blockIdx.x;       // 0..383
  const int c    = bid >> 7;              // channel 0..2
  const int rg   = bid & 127;             // row group
  const int row0 = rg * 32;

  const float* src = (c == 0) ? amp : (c == 1) ? pitch : boundary;
  const float d = decay[c];

  const unsigned shbase = (unsigned)(size_t)(&sh[0]); // low 32b = LDS offset

  // prime the ring with 3 chunks (48 async ops in flight)
  issue_chunk_async(src, row0, lane, shbase, 0);
  issue_chunk_async(src, row0, lane, shbase, 1);
  issue_chunk_async(src, row0, lane, shbase, 2);

  float v = 0.0f;
  const unsigned out_base =
      ((unsigned)c * NB + (unsigned)(row0 + lane)) * WPR;

  for (int k = 0; k < NCHK; ++k) {
    // keep 3 chunks ahead; wait so that chunk k is guaranteed complete
    // (ASYNCcnt completes in order; <=16*(chunks newer than k) outstanding)
    if (k + 3 < NCHK) {
      issue_chunk_async(src, row0, lane, shbase, k + 3);
      asm volatile("s_wait_asynccnt 0x30" ::: "memory");
    } else if (k + 2 < NCHK) {
      asm volatile("s_wait_asynccnt 0x20" ::: "memory");
    } else if (k + 1 < NCHK) {
      asm volatile("s_wait_asynccnt 0x10" ::: "memory");
    } else {
      asm volatile("s_wait_asynccnt 0x0" ::: "memory");
    }

    const float* bufp = &sh[(k & (NBUF - 1)) * (32 * CHK)];
    unsigned w = 0;
#pragma unroll 8
    for (int b = 0; b < CHK / 4; ++b) {       // 16 blocks of 4 steps
      const int pos = (b + lane) & 15;        // undo rotation swizzle
      const float4 x4 =
          *(const float4*)(bufp + lane * CHK + pos * 4);
      const float xs0 = x4.x, xs1 = x4.y, xs2 = x4.z, xs3 = x4.w;

      unsigned s;
      v = fmaf(d, v, xs0); s = (v >= 1.0f); v -= (float)s; w |= s << (((b << 2) + 0) & 31);
      v = fmaf(d, v, xs1); s = (v >= 1.0f); v -= (float)s; w |= s << (((b << 2) + 1) & 31);
      v = fmaf(d, v, xs2); s = (v >= 1.0f); v -= (float)s; w |= s << (((b << 2) + 2) & 31);
      v = fmaf(d, v, xs3); s = (v >= 1.0f); v -= (float)s; w |= s << (((b << 2) + 3) & 31);

      if ((b & 7) == 7) {                     // every 32 steps flush one word
        spk[out_base + (unsigned)(k * (CHK / 32) + (b >> 3))] = w;
        w = 0;
      }
    }
  }
}

// ---------------------------------------------------------------------------
// Kernel 2: per-row sal, row-max normalize, top-5 (lowest index on ties), mu.
// One 256-thread block per row. ds_max_u32/u64 shared atomics for reductions.
// ---------------------------------------------------------------------------

__global__ void __launch_bounds__(256)
sal_topk_kernel(const unsigned* __restrict__ spk,
                const float* __restrict__ weights,
                float* __restrict__ out) {
  __shared__ float sal[NT]; // 32 KB, accessed through SWZ()
  __shared__ unsigned maxbits;
  __shared__ unsigned long long bestkey;
  __shared__ int picked[8];
  __shared__ float pickedraw[8];

  const int row = (int)blockIdx.x;
  const int tid = (int)threadIdx.x;
  const float w0 = weights[0], w1 = weights[1], w2 = weights[2];

  if (tid == 0) maxbits = 0u;
  __syncthreads();

  const unsigned pa = spk[((size_t)0 * NB + row) * WPR + tid];
  const unsigned pp = spk[((size_t)1 * NB + row) * WPR + tid];
  const unsigned pb = spk[((size_t)2 * NB + row) * WPR + tid];

  float lmax = 0.0f;
#pragma unroll 8
  for (int j = 0; j < 32; ++j) {
    const float s = w0 * (float)((pa >> j) & 1u) +
                    w1 * (float)((pp >> j) & 1u) +
                    w2 * (float)((pb >> j) & 1u);
    sal[SWZ(tid * 32 + j)] = s;
    lmax = fmaxf(lmax, s);
  }
  atomicMax(&maxbits, __float_as_uint(lmax)); // sal >= 0 -> bits monotonic
  __syncthreads();

  const float mx  = __uint_as_float(maxbits);
  const float inv = 1.0f / (mx + 1e-6f);

  // normalized sal, fully coalesced across the block
  float* osal = out + NB + (size_t)row * NT;
#pragma unroll 4
  for (int j = 0; j < 32; ++j) {
    const int t = j * 256 + tid;
    osal[t] = sal[SWZ(t)] * inv;
  }

  // top-5: key = (value_bits << 32) | (NT-1-t)  => max picks largest value,
  // ties broken by smallest index (matches jax.lax.top_k ordering).
  for (int k = 0; k < 5; ++k) {
    if (tid == 0) bestkey = 0ull;
    __syncthreads();
    unsigned long long lbest = 0ull;
    for (int j = 0; j < 32; ++j) {
      const int t = tid * 32 + j;
      bool skip = false;
      for (int q = 0; q < k; ++q) skip = skip || (picked[q] == t);
      if (!skip) {
        const unsigned long long key =
            ((unsigned long long)__float_as_uint(sal[SWZ(t)]) << 32) |
            (unsigned)(NT - 1 - t);
        lbest = (key > lbest) ? key : lbest;
      }
    }
    atomicMax(&bestkey, lbest); // ds_max_u64 on LDS
    __syncthreads();
    if (tid == 0) {
      const unsigned long long bk = bestkey;
      picked[k]    = (int)(NT - 1 - (unsigned)(bk & 0xffffffffu));
      pickedraw[k] = __uint_as_float((unsigned)(bk >> 32));
    }
    __syncthreads();
  }

  if (tid == 0) {
    float s = 0.0f;
    for (int k = 0; k < 5; ++k) s += pickedraw[k] * inv;
    const float avg = s * 0.2f;
    const float mu  = 0.5f + 2.0f * tanhf(1.8f * avg);
    out[row] = mu;
    float* oidx = out + NB + (size_t)NB * NT + (size_t)row * 5;
    for (int k = 0; k < 5; ++k) oidx[k] = (float)picked[k];
  }
}

// ---------------------------------------------------------------------------

extern "C" void kernel_launch(void* const* d_in, const int* in_sizes, int n_in,
                              void* d_out, int out_size, void* d_ws,
                              size_t ws_size, hipStream_t stream) {
  (void)in_sizes; (void)n_in; (void)out_size; (void)ws_size;
  const float* amp      = (const float*)d_in[0];
  const float* pitch    = (const float*)d_in[1];
  const float* boundary = (const float*)d_in[2];
  const float* decay    = (const float*)d_in[3];
  const float* weights  = (const float*)d_in[4];
  float* out = (float*)d_out;
  unsigned* spk = (unsigned*)d_ws; // 3 * 4096 * 256 * 4B = 12.6 MB packed spikes

  // 3 channels * (4096/32) row groups = 384 one-wave workgroups
  lif_scan_kernel<<<dim3(384), dim3(32), 0, stream>>>(amp, pitch, boundary,
                                                      decay, spk);
  // one block per row
  sal_topk_kernel<<<dim3(NB), dim3(256), 0, stream>>>(spk, weights, out);
}